// LE_LOSS_34299608826565
// MI455X (gfx1250) — compile-verified
//
#include <hip/hip_runtime.h>
#include <math.h>

typedef float v2f __attribute__((ext_vector_type(2)));
typedef float v8f __attribute__((ext_vector_type(8)));

#define N_PATCH 8192   // B * (H/s)^2 = 32*16*16
#define DIMS    256    // s*s
#define KNN     10
#define B_SZ    32
#define IMG     256
#define PS      16
#define ASTR    258    // LDS row stride (floats) to avoid 16-way bank conflicts

// ---------------------------------------------------------------------------
// Patchify: (B,1,H,W) -> (N, 256) row-major patches, scaled by 1/100
// ---------------------------------------------------------------------------
__global__ void patchify_kernel(const float* __restrict__ src, float* __restrict__ dst) {
  int tid = blockIdx.x * blockDim.x + threadIdx.x;
  if (tid >= N_PATCH * DIMS) return;
  int row = tid >> 8;            // patch id
  int col = tid & 255;           // element within patch
  int b   = row >> 8;            // row / 256 patches-per-image
  int rem = row & 255;
  int m = rem >> 4, n = rem & 15;
  int r = col >> 4, c = col & 15;
  int src_idx = (b * IMG + m * PS + r) * IMG + n * PS + c;
  dst[tid] = src[src_idx] * 0.01f;
}

// ---------------------------------------------------------------------------
// Row squared norms: one wave per row
// ---------------------------------------------------------------------------
__global__ void rownorm_kernel(const float* __restrict__ mv, float* __restrict__ sq) {
  int gwave = (blockIdx.x * blockDim.x + threadIdx.x) >> 5;
  int lane  = threadIdx.x & 31;
  if (gwave >= N_PATCH) return;
  const float* rowp = mv + (size_t)gwave * DIMS;
  float s = 0.f;
  for (int t = lane; t < DIMS; t += 32) { float v = rowp[t]; s += v * v; }
  for (int off = 16; off; off >>= 1) s += __shfl_xor(s, off, 32);
  if (lane == 0) sq[gwave] = s;
}

// ---------------------------------------------------------------------------
// MSE partial reduction into acc[0]
// ---------------------------------------------------------------------------
__global__ void mse_kernel(const float* __restrict__ a, const float* __restrict__ b,
                           float* __restrict__ acc, int n) {
  __shared__ float red[8];
  int tid = blockIdx.x * blockDim.x + threadIdx.x;
  float s = 0.f;
  for (int i = tid; i < n; i += gridDim.x * blockDim.x) {
    float d = a[i] - b[i];
    s += d * d;
  }
  for (int off = 16; off; off >>= 1) s += __shfl_xor(s, off, 32);
  int lane = threadIdx.x & 31, w = threadIdx.x >> 5;
  if (lane == 0) red[w] = s;
  __syncthreads();
  if (threadIdx.x == 0) {
    float t = 0.f;
    for (int q = 0; q < 8; ++q) t += red[q];
    atomicAdd(&acc[0], t);
  }
}

__global__ void init_kernel(float* acc) {
  if (threadIdx.x < 2) acc[threadIdx.x] = 0.f;
}

// ---------------------------------------------------------------------------
// kNN: Gram matrix via V_WMMA_F32_16X16X4_F32 + streaming top-10 per row.
// One workgroup (8 waves) per 16-row tile; each wave strides column tiles.
// ---------------------------------------------------------------------------
__global__ __launch_bounds__(256) void knn_kernel(const float* __restrict__ mv,
                                                  const float* __restrict__ sq,
                                                  int* __restrict__ topk) {
  __shared__ float Alds[16 * ASTR];
  __shared__ float sqa[16];
  __shared__ float Dt[8][16][17];       // per-wave distance tile (transposed access)
  __shared__ float Md[8][16][KNN];      // per-wave top-k merge buffers
  __shared__ int   Mi[8][16][KNN];

  const int mbase = blockIdx.x * 16;
  const int lane  = threadIdx.x & 31;
  const int wv    = threadIdx.x >> 5;

  // Stage A tile (16 rows x 256 dims) into LDS
  for (int i = threadIdx.x; i < 16 * DIMS; i += 256) {
    int r = i >> 8, c = i & 255;
    Alds[r * ASTR + c] = mv[(size_t)(mbase + r) * DIMS + c];
  }
  if (threadIdx.x < 16) sqa[threadIdx.x] = sq[mbase + threadIdx.x];
  __syncthreads();

  // Register-resident sorted (ascending) top-10 list per owner lane (lane<16)
  float bd[KNN]; int bi[KNN];
#pragma unroll
  for (int j = 0; j < KNN; ++j) { bd[j] = 3.4e38f; bi[j] = 0; }

  const int r  = lane & 15;            // row within A-frag / col within B-frag
  const int kh = (lane >> 4) << 1;     // K sub-offset: lanes 16-31 hold K=2,3
  const int mh = (lane >> 4) * 8;      // C/D: upper lanes hold M=8..15

  const float* abase = &Alds[r * ASTR + kh];

  for (int ct = wv; ct < N_PATCH / 16; ct += 8) {
    const int nbase = ct * 16;
    const float* bbase = mv + (size_t)(nbase + r) * DIMS + kh;

    v8f acc = {};
#pragma unroll 4
    for (int k0 = 0; k0 < DIMS; k0 += 4) {
      v2f av, bv;
      av.x = abase[k0];     av.y = abase[k0 + 1];
      bv.x = bbase[k0];     bv.y = bbase[k0 + 1];
      // D = A(16x4,f32) * B(4x16,f32) + C ; full f32 precision matrix op
      acc = __builtin_amdgcn_wmma_f32_16x16x4_f32(
          false, av, false, bv, (short)0, acc, false, false);
    }

    // dist[m][n] = sq[m] + sq[n] - 2*dot ; lane holds column n=r, rows mh..mh+7
    const float sqn = sq[nbase + r];
    asm volatile("" ::: "memory");
#pragma unroll
    for (int v = 0; v < 8; ++v) {
      Dt[wv][mh + v][r] = sqa[mh + v] + sqn - 2.0f * acc[v];
    }
    asm volatile("" ::: "memory");   // LDS ops are in-order per wave

    if (lane < 16) {
#pragma unroll
      for (int n = 0; n < 16; ++n) {
        float d = Dt[wv][lane][n];
        if (d < bd[KNN - 1]) {
          bd[KNN - 1] = d; bi[KNN - 1] = nbase + n;
#pragma unroll
          for (int j = KNN - 1; j > 0; --j) {
            if (bd[j] < bd[j - 1]) {
              float td = bd[j]; bd[j] = bd[j - 1]; bd[j - 1] = td;
              int   ti = bi[j]; bi[j] = bi[j - 1]; bi[j - 1] = ti;
            }
          }
        }
      }
    }
    asm volatile("" ::: "memory");
  }

  // Merge 8 per-wave lists through LDS
  if (lane < 16) {
#pragma unroll
    for (int j = 0; j < KNN; ++j) { Md[wv][lane][j] = bd[j]; Mi[wv][lane][j] = bi[j]; }
  }
  __syncthreads();

  if (threadIdx.x < 16) {
    const int row = threadIdx.x;
    float fd[KNN]; int fi[KNN];
#pragma unroll
    for (int j = 0; j < KNN; ++j) { fd[j] = 3.4e38f; fi[j] = 0; }
    for (int w = 0; w < 8; ++w) {
      for (int j = 0; j < KNN; ++j) {
        float d = Md[w][row][j]; int id = Mi[w][row][j];
        if (d < fd[KNN - 1]) {
          fd[KNN - 1] = d; fi[KNN - 1] = id;
#pragma unroll
          for (int q = KNN - 1; q > 0; --q) {
            if (fd[q] < fd[q - 1]) {
              float td = fd[q]; fd[q] = fd[q - 1]; fd[q - 1] = td;
              int   ti = fi[q]; fi[q] = fi[q - 1]; fi[q - 1] = ti;
            }
          }
        }
      }
    }
#pragma unroll
    for (int j = 0; j < KNN; ++j) topk[(mbase + row) * KNN + j] = fi[j];
  }
}

// ---------------------------------------------------------------------------
// Pair loss: one wave per (row, neighbor) pair; sum pd*w into acc[1]
// ---------------------------------------------------------------------------
__global__ __launch_bounds__(256) void pair_kernel(const float* __restrict__ mv,
                                                   const float* __restrict__ pv,
                                                   const int* __restrict__ topk,
                                                   float* __restrict__ acc) {
  __shared__ float red[8];
  int gw   = (blockIdx.x * blockDim.x + threadIdx.x) >> 5;  // pair id
  int lane = threadIdx.x & 31;
  float contrib = 0.f;
  if (gw < N_PATCH * KNN) {
    int i = gw / KNN;
    int j = topk[gw];
    const float* mi_ = mv + (size_t)i * DIMS;
    const float* mj_ = mv + (size_t)j * DIMS;
    const float* pi_ = pv + (size_t)i * DIMS;
    const float* pj_ = pv + (size_t)j * DIMS;
    float s2 = 0.f, s1 = 0.f, p2 = 0.f;
    for (int t = lane; t < DIMS; t += 32) {
      float dm = mi_[t] - mj_[t];
      s2 += dm * dm;
      s1 += fabsf(dm);
      float dp = pi_[t] - pj_[t];
      p2 += dp * dp;
    }
    for (int off = 16; off; off >>= 1) {
      s2 += __shfl_xor(s2, off, 32);
      s1 += __shfl_xor(s1, off, 32);
      p2 += __shfl_xor(p2, off, 32);
    }
    if (lane == 0) {
      float dl2 = sqrtf(s2);
      float dl1 = s1 + 1e-8f;
      float w   = expf(-dl2 / dl1);
      float pd  = sqrtf(fmaxf(p2, 0.f));
      contrib = pd * w;
    }
  }
  int w_ = threadIdx.x >> 5;
  if (lane == 0) red[w_] = contrib;
  __syncthreads();
  if (threadIdx.x == 0) {
    float t = 0.f;
    for (int q = 0; q < 8; ++q) t += red[q];
    atomicAdd(&acc[1], t);
  }
}

__global__ void final_kernel(const float* __restrict__ acc,
                             const float* __restrict__ lamda,
                             float* __restrict__ out) {
  if (threadIdx.x == 0 && blockIdx.x == 0) {
    float mse   = acc[0] / (float)(B_SZ * IMG * IMG);
    float loss2 = acc[1] / ((float)N_PATCH * (float)N_PATCH);
    out[0] = mse + loss2 * lamda[0];
  }
}

// ---------------------------------------------------------------------------
extern "C" void kernel_launch(void* const* d_in, const int* in_sizes, int n_in,
                              void* d_out, int out_size, void* d_ws, size_t ws_size,
                              hipStream_t stream) {
  const float* pred  = (const float*)d_in[0];
  const float* maps  = (const float*)d_in[1];
  const float* lamda = (const float*)d_in[3];

  char*  ws   = (char*)d_ws;
  float* mv   = (float*)(ws);                                  // 8 MB
  float* pv   = (float*)(ws + 8u * 1024 * 1024);               // 8 MB
  float* sq   = (float*)(ws + 16u * 1024 * 1024);              // 32 KB
  int*   topk = (int*)  (ws + 16u * 1024 * 1024 + 64 * 1024);  // 320 KB
  float* acc  = (float*)(ws + 16u * 1024 * 1024 + 512 * 1024); // 2 floats
  float* out  = (float*)d_out;

  const int nel = N_PATCH * DIMS;  // 2,097,152

  init_kernel<<<1, 32, 0, stream>>>(acc);
  patchify_kernel<<<nel / 256, 256, 0, stream>>>(maps, mv);
  patchify_kernel<<<nel / 256, 256, 0, stream>>>(pred, pv);
  rownorm_kernel<<<(N_PATCH * 32) / 256, 256, 0, stream>>>(mv, sq);
  mse_kernel<<<1024, 256, 0, stream>>>(maps, pred, acc, B_SZ * IMG * IMG);
  knn_kernel<<<N_PATCH / 16, 256, 0, stream>>>(mv, sq, topk);
  pair_kernel<<<(N_PATCH * KNN) / 8, 256, 0, stream>>>(mv, pv, topk, acc);
  final_kernel<<<1, 32, 0, stream>>>(acc, lamda, out);
}